// FixedRadiusNearNeighbors_3324304687804
// MI455X (gfx1250) — compile-verified
//
#include <hip/hip_runtime.h>

typedef __attribute__((ext_vector_type(2))) float v2f;
typedef __attribute__((ext_vector_type(8))) float v8f;

#define FRNN_R2   0.01f     // RADIUS^2
#define FRNN_K    64        // N_NEIGHBOR
#define FRNN_N    16384     // points per batch
#define FRNN_S    2048      // centroids per batch
#define FRNN_B    8         // batches

static __device__ __forceinline__ int imin(int a, int b) { return a < b ? a : b; }

// One wave owns 16 centroids; streams all N points in 16-wide WMMA tiles
// (V_WMMA_F32_16X16X4_F32, A row = (-2cx,-2cy,-2cz,|c|^2), B col = (px,py,pz,1),
// C = inline 0, pass iff D <= R^2 - |p|^2), compacting qualifying indices in
// ascending order via ballot/popcount. 2 tiles/iter + 2-deep load pipeline.
__global__ void __launch_bounds__(32)
frnn_wmma_kernel(const float* __restrict__ pos,
                 const int*   __restrict__ cent,
                 int*         __restrict__ out)
{
    const int lane = threadIdx.x & 31;
    const int half = lane >> 4;            // 0 = lanes 0-15, 1 = lanes 16-31
    const int l16  = lane & 15;
    const unsigned laneLT   = (1u << l16) - 1u;
    // loop-invariant: bits of MY row's ballot half strictly below my column
    const unsigned rankMask = half ? (laneLT << 16) : laneLT;

    const int wid = blockIdx.x;            // one wave per block
    const int b   = wid / (FRNN_S / 16);
    const int s0  = (wid % (FRNN_S / 16)) * 16;

    const float* posB = pos  + (size_t)b * FRNN_N * 3;
    const int*   cenB = cent + (size_t)b * FRNN_S;
    int*         outB = out  + ((size_t)b * FRNN_S + s0) * FRNN_K;

    // ---- A matrix (16x4 f32): row m = (-2cx, -2cy, -2cz, |c|^2) ----
    // ISA layout: lane l<16 holds (K0,K1), lane l+16 holds (K2,K3).
    const int cidx = cenB[s0 + l16];
    const float cx = posB[cidx * 3 + 0];
    const float cy = posB[cidx * 3 + 1];
    const float cz = posB[cidx * 3 + 2];
    v2f A;
    if (half == 0) { A.x = -2.0f * cx; A.y = -2.0f * cy; }
    else           { A.x = -2.0f * cz; A.y = cx*cx + cy*cy + cz*cz; }

    const v8f Czero = {};                  // folds to WMMA inline-0 SRC2

    // Wave-uniform counters (ballot-derived) -> SGPRs; constant indices only
    int cntLo[8], cntHi[8];
#pragma unroll
    for (int v = 0; v < 8; ++v) { cntLo[v] = 0; cntHi[v] = 0; }

#define FRNN_LOAD(tt, X, Y, Z) do {                                           \
        const float* _p = posB + (size_t)((tt) + l16) * 3; /* global_load_b96 */ \
        (X) = _p[0]; (Y) = _p[1]; (Z) = _p[2];                                \
    } while (0)

#define FRNN_TILE(tt, PX, PY, PZ) do {                                        \
        const float _sq  = (PX)*(PX) + (PY)*(PY) + (PZ)*(PZ);                 \
        const float _thr = FRNN_R2 - _sq;   /* |p|^2 folded into threshold */ \
        v2f _Bm;                            /* B col n = (px, py, pz, 1)   */ \
        _Bm.x = half ? (PZ) : (PX);                                           \
        _Bm.y = half ? 1.0f : (PY);                                           \
        v8f _D = __builtin_amdgcn_wmma_f32_16x16x4_f32(                       \
            false, A, false, _Bm, (short)0, Czero, false, false);             \
        const int _n = (tt) + l16;                                            \
        _Pragma("unroll")                                                     \
        for (int v = 0; v < 8; ++v) {                                         \
            const bool _pass = _D[v] <= _thr;                                 \
            const unsigned _mask = __builtin_amdgcn_ballot_w32(_pass);        \
            const unsigned _slot =                                            \
                (unsigned)(half ? cntHi[v] : cntLo[v]) +                      \
                (unsigned)__builtin_popcount(_mask & rankMask);               \
            if (_pass && _slot < FRNN_K) {        /* single saveexec block */ \
                outB[(unsigned)v * FRNN_K + ((unsigned)half << 9) + _slot] = _n; \
            }                                                                 \
            cntLo[v] += __builtin_popcount(_mask & 0xFFFFu);                  \
            cntHi[v] += __builtin_popcount(_mask >> 16);                      \
        }                                                                     \
    } while (0)

    // software pipeline: 2 tiles in flight
    float x0,y0,z0, x1,y1,z1, nx0,ny0,nz0, nx1,ny1,nz1;
    FRNN_LOAD(0,  x0, y0, z0);
    FRNN_LOAD(16, x1, y1, z1);

    for (int t = 0; t < FRNN_N; t += 32) {
        const int tn0 = imin(t + 32, FRNN_N - 16);   // clamped prefetch
        const int tn1 = imin(t + 48, FRNN_N - 16);
        FRNN_LOAD(tn0, nx0, ny0, nz0);
        FRNN_LOAD(tn1, nx1, ny1, nz1);

        FRNN_TILE(t,      x0, y0, z0);
        FRNN_TILE(t + 16, x1, y1, z1);

        x0 = nx0; y0 = ny0; z0 = nz0;
        x1 = nx1; y1 = ny1; z1 = nz1;
    }

#undef FRNN_LOAD
#undef FRNN_TILE

    // Padding: slots [cnt, 64) <- first hit. The first hit (if any) is already
    // at slot 0 in memory (written this scan; same-wave store->load to the
    // same address is ordered), so read it back; empty rows fill with N,
    // matching the reference's where(idx==N, group_first) semantics.
#pragma unroll
    for (int v = 0; v < 8; ++v) {
        const int cnt = half ? cntHi[v] : cntLo[v];
        const unsigned rowOff = (unsigned)v * FRNN_K + ((unsigned)half << 9);
        const int fill = (cnt > 0) ? outB[rowOff] : FRNN_N;
#pragma unroll
        for (int k = 0; k < 4; ++k) {
            const int slot = l16 + k * 16;       // 16 lanes x 4 cover 64 slots
            if (slot >= cnt) outB[rowOff + (unsigned)slot] = fill;
        }
    }
}

extern "C" void kernel_launch(void* const* d_in, const int* in_sizes, int n_in,
                              void* d_out, int out_size, void* d_ws, size_t ws_size,
                              hipStream_t stream) {
    (void)in_sizes; (void)n_in; (void)out_size; (void)d_ws; (void)ws_size;
    const float* pos  = (const float*)d_in[0];   // (B, N, 3) f32
    const int*   cent = (const int*)d_in[1];     // (B, S)    i32
    int*         out  = (int*)d_out;             // (B, S, 64) i32

    const int nWaves = FRNN_B * (FRNN_S / 16);   // 1024 waves, one per block
    frnn_wmma_kernel<<<nWaves, 32, 0, stream>>>(pos, cent, out);
}